// ERA5ToModelConverter_33320356282815
// MI455X (gfx1250) — compile-verified
//
#include <hip/hip_runtime.h>
#include <math.h>
#include <stdint.h>

// Problem constants (match reference)
constexpr int HE_ = 361, WE_ = 720, L_ = 37;
constexpr int HM_ = 180, WM_ = 360, K_ = 32;
constexpr float EPS_ = 1e-8f;
constexpr int NPTS_ = HM_ * WM_;          // 64800 model columns
constexpr int PTS_PER_BLOCK_ = 8;         // 8 wave32 per 256-thread block

// searchsorted(..., side='right'): count of g[i] <= pt
__device__ __forceinline__ int ss_right(const float* g, int n, float pt) {
  int lo = 0, hi = n;
  while (lo < hi) {
    int mid = (lo + hi) >> 1;
    if (g[mid] <= pt) lo = mid + 1; else hi = mid;
  }
  return lo;
}

// gfx1250 async global->LDS DMA: one B32 element per active lane.
// VDST operand = VGPR holding LDS byte offset; VADDR = VGPR pair with the
// 64-bit global address (GV mode). Tracked by ASYNCcnt.
__device__ __forceinline__ void async_g2lds_b32(uint32_t lds_byte_off,
                                                const float* gptr) {
  uint64_t ga = (uint64_t)(uintptr_t)gptr;
  asm volatile("global_load_async_to_lds_b32 %0, %1, off"
               :
               : "v"(lds_byte_off), "v"(ga)
               : "memory");
}

__device__ __forceinline__ void wait_asynccnt0() {
  asm volatile("s_wait_asynccnt 0x0" ::: "memory");
}

__global__ __launch_bounds__(256, 4)
void era5_regrid_kernel(
    const float* __restrict__ gu, const float* __restrict__ gv,
    const float* __restrict__ gw, const float* __restrict__ gT,
    const float* __restrict__ gq, const float* __restrict__ gps,
    const float* __restrict__ era5_lat, const float* __restrict__ era5_lon,
    const float* __restrict__ model_lat, const float* __restrict__ model_lon,
    const float* __restrict__ p_levels, const float* __restrict__ a_k,
    const float* __restrict__ b_k,
    float* __restrict__ out3d,      // (HM, WM, K, 5)
    float* __restrict__ pmod_out,   // (HM, WM, K)
    float* __restrict__ ps_out)     // (HM, WM)
{
  // Small lookup tables in LDS (320 KB/WGP -> trivial footprint, kills
  // dependent-global-load latency in the binary searches).
  __shared__ float s_lat[HE_];
  __shared__ float s_lon[WE_];
  __shared__ float s_plev[L_];
  __shared__ float s_logpe[L_];

  const int tid = threadIdx.x;

  // Stage lat/lon tables with the CDNA5 async memory->LDS path (no VGPR
  // round-trip; ASYNCcnt-tracked). 361 + 720 elements, stride-256 per thread.
  for (int i = tid; i < HE_; i += 256)
    async_g2lds_b32((uint32_t)(uintptr_t)(&s_lat[i]), era5_lat + i);
  for (int i = tid; i < WE_; i += 256)
    async_g2lds_b32((uint32_t)(uintptr_t)(&s_lon[i]), era5_lon + i);

  // Pressure tables need a logf per element -> normal VALU path.
  if (tid < L_) {
    float pl = p_levels[tid];
    s_plev[tid]  = pl;
    s_logpe[tid] = logf(pl + EPS_);
  }

  wait_asynccnt0();   // async LDS writes landed for this wave
  __syncthreads();    // ...and are visible block-wide

  const int wave = tid >> 5;      // wave32: one wave per model column
  const int k    = tid & 31;      // lane = output model level (K == 32)
  const int p    = blockIdx.x * PTS_PER_BLOCK_ + wave;
  if (p >= NPTS_) return;
  const int y = p / WM_;
  const int x = p - y * WM_;

  // ---- horizontal interp indices/weights (wave-uniform) ----
  const float mlat = model_lat[y];
  const float mlon = model_lon[x];
  int iy = ss_right(s_lat, HE_, mlat) - 1;
  iy = iy < 0 ? 0 : (iy > HE_ - 2 ? HE_ - 2 : iy);
  int ix = ss_right(s_lon, WE_, mlon) - 1;
  ix = ix < 0 ? 0 : (ix > WE_ - 2 ? WE_ - 2 : ix);
  const float ty  = (mlat - s_lat[iy]) / (s_lat[iy + 1] - s_lat[iy]);
  const float tx  = (mlon - s_lon[ix]) / (s_lon[ix + 1] - s_lon[ix]);
  const float omy = 1.0f - ty, omx = 1.0f - tx;

  // Corner column bases (contiguous 37-level profiles)
  const int c00 = (iy * WE_ + ix) * L_;
  const int c01 = c00 + L_;
  const int c10 = c00 + WE_ * L_;
  const int c11 = c10 + L_;

  const float* const vars[5] = {gu, gv, gw, gT, gq};

  // gfx1250 prefetch (global_prefetch_b8): pull the four corner profiles
  // (~148 B each; c00/c01 and c10/c11 are adjacent) toward the WGP while the
  // lane-level binary searches + logf below are in flight. Wave-uniform
  // addresses -> one request per prefetch.
#pragma unroll
  for (int vi = 0; vi < 5; ++vi) {
    const float* A = vars[vi];
    __builtin_prefetch(A + c00,      0, 0);
    __builtin_prefetch(A + c00 + 32, 0, 0);
    __builtin_prefetch(A + c00 + 64, 0, 0);
    __builtin_prefetch(A + c10,      0, 0);
    __builtin_prefetch(A + c10 + 32, 0, 0);
    __builtin_prefetch(A + c10 + 64, 0, 0);
  }

  // ---- surface pressure bilinear (wave-uniform) ----
  const int s00 = iy * WE_ + ix;
  const float p00 = gps[s00], p01 = gps[s00 + 1];
  const float p10 = gps[s00 + WE_], p11 = gps[s00 + WE_ + 1];
  const float ps_m = omy * (omx * p00 + tx * p01) + ty * (omx * p10 + tx * p11);

  const int nvalid = ss_right(s_plev, L_, ps_m);

  // ---- vertical interp (per lane = per model level) ----
  const float pmod  = a_k[k] + b_k[k] * ps_m;
  const float logpm = logf(pmod + EPS_);
  int idx = ss_right(s_logpe, L_, logpm) - 1;
  int upper = nvalid - 2; if (upper < 0) upper = 0;
  idx = idx < 0 ? 0 : (idx > upper ? upper : idx);

  const float x0 = s_logpe[idx];
  const float x1 = s_logpe[idx + 1];
  const float t  = (logpm - x0) / (x1 - x0);
  const bool valid = (nvalid >= 2);

  const size_t outbase = ((size_t)p * K_ + k) * 5;
#pragma unroll
  for (int vi = 0; vi < 5; ++vi) {
    const float* A = vars[vi];
    // Gather the two needed levels from each corner profile (adjacent floats)
    const float a00 = A[c00 + idx], b00 = A[c00 + idx + 1];
    const float a01 = A[c01 + idx], b01 = A[c01 + idx + 1];
    const float a10 = A[c10 + idx], b10 = A[c10 + idx + 1];
    const float a11 = A[c11 + idx], b11 = A[c11 + idx + 1];
    // Same nesting as the reference's hd expression (interp commutes per level)
    const float g0 = omy * (omx * a00 + tx * a01) + ty * (omx * a10 + tx * a11);
    const float g1 = omy * (omx * b00 + tx * b01) + ty * (omx * b10 + tx * b11);
    float o = g0 + t * (g1 - g0);
    if (!valid) o = 0.0f;                                   // mask BEFORE clip
    if (vi == 3) o = fminf(fmaxf(o, 150.0f), 350.0f);       // T clip
    if (vi == 4) o = fminf(fmaxf(o, 0.0f), 0.05f);          // q clip
    out3d[outbase + vi] = o;
  }

  pmod_out[(size_t)p * K_ + k] = pmod;   // p_model is unmasked/unclipped
  if (k == 0) ps_out[p] = fminf(fmaxf(ps_m, 30000.0f), 110000.0f);
}

extern "C" void kernel_launch(void* const* d_in, const int* in_sizes, int n_in,
                              void* d_out, int out_size, void* d_ws, size_t ws_size,
                              hipStream_t stream) {
  (void)in_sizes; (void)n_in; (void)out_size; (void)d_ws; (void)ws_size;
  const float* gu        = (const float*)d_in[0];
  const float* gv        = (const float*)d_in[1];
  const float* gw        = (const float*)d_in[2];
  const float* gT        = (const float*)d_in[3];
  const float* gq        = (const float*)d_in[4];
  const float* gps       = (const float*)d_in[5];
  const float* era5_lat  = (const float*)d_in[6];
  const float* era5_lon  = (const float*)d_in[7];
  const float* model_lat = (const float*)d_in[8];
  const float* model_lon = (const float*)d_in[9];
  const float* p_levels  = (const float*)d_in[10];
  const float* a_k       = (const float*)d_in[11];
  const float* b_k       = (const float*)d_in[12];

  float* out      = (float*)d_out;
  float* out3d    = out;                                  // HM*WM*K*5
  float* pmod_out = out3d + (size_t)HM_ * WM_ * K_ * 5;   // HM*WM*K
  float* ps_out   = pmod_out + (size_t)HM_ * WM_ * K_;    // HM*WM

  const int nblocks = (NPTS_ + PTS_PER_BLOCK_ - 1) / PTS_PER_BLOCK_;  // 8100
  era5_regrid_kernel<<<nblocks, 256, 0, stream>>>(
      gu, gv, gw, gT, gq, gps, era5_lat, era5_lon, model_lat, model_lon,
      p_levels, a_k, b_k, out3d, pmod_out, ps_out);
}